// PolynomialFeatures_45509473468569
// MI455X (gfx1250) — compile-verified
//
#include <hip/hip_runtime.h>
#include <stdint.h>

#define TPB   256
#define ROWS  8
#define XDIM  32
#define XEXT  33   // 32 inputs + constant-1 column (index 32)

#if defined(__HIP_DEVICE_COMPILE__) && defined(__gfx1250__) && \
    __has_builtin(__builtin_amdgcn_global_load_async_to_lds_b32)
#define POLY_USE_ASYNC 1
#endif

__global__ __launch_bounds__(TPB) void PolynomialFeatures_45509473468569_kernel(
    const float* __restrict__ x, const int* __restrict__ idx,
    float* __restrict__ out, int B, int F) {
  __shared__ float xs[ROWS * XEXT];

  const int tid  = threadIdx.x;
  const int row0 = blockIdx.y * ROWS;
  const int f    = blockIdx.x * TPB + tid;

  // ---- Stage ROWS x XDIM tile of x into LDS: 256 elements, 1 per thread ----
  const int r    = tid >> 5;   // row within tile (wave r handles row r)
  const int c    = tid & 31;   // column
  const int grow = row0 + r;

#if defined(POLY_USE_ASYNC)
  // CDNA5 async copy: global -> LDS, 4B per lane, tracked by ASYNCcnt.
  // Builtin signature (from probe diagnostic): (AS1 int*, AS3 int*, imm, imm).
  if (grow < B) {
    typedef __attribute__((address_space(1))) int gbl_i32;
    typedef __attribute__((address_space(3))) int lds_i32;
    __builtin_amdgcn_global_load_async_to_lds_b32(
        (gbl_i32*)(uintptr_t)(x + (size_t)grow * XDIM + c),
        (lds_i32*)(uint32_t)(uintptr_t)(&xs[r * XEXT + c]),
        /*offset=*/0, /*cpol=*/0);
  }
#else
  xs[r * XEXT + c] = (grow < B) ? x[(size_t)grow * XDIM + c] : 0.0f;
#endif
  if (tid < ROWS) xs[tid * XEXT + XDIM] = 1.0f;  // x_ext[:,32] == 1

  // ---- Index triple for this feature; issue before barrier to overlap ----
  int i0 = 0, i1 = 0, i2 = 0;
  if (f < F) {
    __builtin_prefetch(&idx[3 * (size_t)f], 0, 0);  // global_prefetch_b8
    i0 = idx[3 * f + 0];
    i1 = idx[3 * f + 1];
    i2 = idx[3 * f + 2];
  }

#if defined(POLY_USE_ASYNC)
#if __has_builtin(__builtin_amdgcn_s_wait_asynccnt)
  __builtin_amdgcn_s_wait_asynccnt(0);
#else
  asm volatile("s_wait_asynccnt 0" ::: "memory");
#endif
#endif
  __syncthreads();

  // ---- 3-way gather-product, 8 coalesced row outputs per thread ----
  if (f < F) {
#pragma unroll
    for (int rr = 0; rr < ROWS; ++rr) {
      const int gr = row0 + rr;
      if (gr < B) {
        const float* xr = &xs[rr * XEXT];
        out[(size_t)gr * F + f] = xr[i0] * xr[i1] * xr[i2];
      }
    }
  }
}

extern "C" void kernel_launch(void* const* d_in, const int* in_sizes, int n_in,
                              void* d_out, int out_size, void* d_ws, size_t ws_size,
                              hipStream_t stream) {
  (void)n_in; (void)d_ws; (void)ws_size; (void)out_size;
  const float* x  = (const float*)d_in[0];   // (B, 32) f32
  const int* idx  = (const int*)d_in[1];     // (F, 3) i32
  float* out      = (float*)d_out;           // (B, F) f32

  const int B = in_sizes[0] / XDIM;
  const int F = in_sizes[1] / 3;

  dim3 grid((F + TPB - 1) / TPB, (B + ROWS - 1) / ROWS);
  PolynomialFeatures_45509473468569_kernel<<<grid, dim3(TPB), 0, stream>>>(
      x, idx, out, B, F);
}